// BertCRF_82849919140549
// MI455X (gfx1250) — compile-verified
//
#include <hip/hip_runtime.h>
#include <math.h>

typedef __attribute__((ext_vector_type(2))) float v2f;
typedef __attribute__((ext_vector_type(8))) float v8f;

#define B_ 64
#define T_ 512
#define F_ 1536
#define C_ 25

#define KCHUNK 128            // K-chunk of W staged in LDS
#define CPAD   40             // padded N stride: zero cols 25..39; 2*CPAD mod 64
                              //  = 16 banks -> half-waves hit disjoint banks
#define WAVES_PER_BLOCK 8     // 256 threads, each wave owns one 16-row M tile

// ---------------------------------------------------------------------------
// Kernel 1: emissions = features @ W + b   via V_WMMA_F32_16X16X4_F32
// features: (B*T, F) row-major, W: (F, C) row-major, emis: (B*T, C)
// One wave handles a 16-row M tile and both 16-wide N tiles (cols 0-15, 16-31;
// cols 25-31 multiply against zero-padded LDS columns -> exact zeros).
// ---------------------------------------------------------------------------
__global__ __launch_bounds__(256) void emis_gemm_kernel(
    const float* __restrict__ feat, const float* __restrict__ W,
    const float* __restrict__ bias, float* __restrict__ emis) {
  __shared__ float wlds[KCHUNK * CPAD];   // 20 KB

  const int lane = threadIdx.x & 31;
  const int wave = threadIdx.x >> 5;
  const int tile = blockIdx.x * WAVES_PER_BLOCK + wave;  // M tile index
  const long row0 = (long)tile * 16;

  // f32 16x16x4 A layout: lanes 0-15 -> K={0,1}, lanes 16-31 -> K={2,3}; M = lane&15
  const int m  = lane & 15;
  const int kb = (lane >> 4) * 2;
  const int n  = lane & 15;      // B/C/D column within N tile
  const int n2 = n + 16;         // second N tile column (valid only if < 25)

  v8f c0 = {};   // cols 0..15
  v8f c1 = {};   // cols 16..31 (only 16..24 stored)

  const float* arow = feat + (row0 + m) * (long)F_;

  // Zero the padded staging buffer once (pad columns stay zero forever).
  for (int i = threadIdx.x; i < KCHUNK * CPAD; i += 256) wlds[i] = 0.0f;

  for (int kc = 0; kc < F_; kc += KCHUNK) {
    __syncthreads();   // previous chunk fully consumed (also covers zero-fill)
    for (int i = threadIdx.x; i < KCHUNK * C_; i += 256) {
      const int kk = i / C_;
      const int nn = i - kk * C_;
      wlds[kk * CPAD + nn] = W[kc * C_ + i];
    }
    __syncthreads();

    // Prefetch this lane's next feature chunk while we compute on this one.
    if (kc + KCHUNK < F_)
      __builtin_prefetch(&arow[kc + KCHUNK + kb], 0, 0);

    for (int k = 0; k < KCHUNK; k += 4) {
      v2f a;
      a.x = arow[kc + k + kb];
      a.y = arow[kc + k + kb + 1];

      v2f b0, b1;
      b0.x = wlds[(k + kb) * CPAD + n];
      b0.y = wlds[(k + kb + 1) * CPAD + n];
      b1.x = wlds[(k + kb) * CPAD + n2];
      b1.y = wlds[(k + kb + 1) * CPAD + n2];

      c0 = __builtin_amdgcn_wmma_f32_16x16x4_f32(false, a, false, b0,
                                                 (short)0, c0, false, false);
      c1 = __builtin_amdgcn_wmma_f32_16x16x4_f32(false, a, false, b1,
                                                 (short)0, c1, false, false);
    }
  }

  // D layout: VGPR v -> row M = v (lanes 0-15) or v+8 (lanes 16-31); N = lane&15
  const float bn  = bias[n];
  const float bn2 = (n2 < C_) ? bias[n2] : 0.0f;
#pragma unroll
  for (int v = 0; v < 8; ++v) {
    const int row = (lane < 16) ? v : (v + 8);
    emis[(row0 + row) * C_ + n] = c0[v] + bn;
    if (n2 < C_)
      emis[(row0 + row) * C_ + n2] = c1[v] + bn2;
  }
}

// ---------------------------------------------------------------------------
// Kernel 2: CRF forward (log-partition) + gold score, one wave per sequence.
// Lane j owns tag state j (j<25). Alpha broadcast via lane shuffles.
// ---------------------------------------------------------------------------
__global__ __launch_bounds__(32) void crf_fwd_kernel(
    const float* __restrict__ emis, const float* __restrict__ trans,
    const float* __restrict__ start_t, const float* __restrict__ end_t,
    const int* __restrict__ lens, const int* __restrict__ labels,
    float* __restrict__ loss_b) {
  const int b = blockIdx.x;
  const int j = threadIdx.x;
  const bool valid = (j < C_);
  const float NEG = -1e30f;

  int len = lens[b];
  len = len < 1 ? 1 : (len > T_ ? T_ : len);

  // Transition column trans[:, j] in registers.
  float tcol[C_];
#pragma unroll
  for (int i = 0; i < C_; ++i) tcol[i] = valid ? trans[i * C_ + j] : 0.0f;

  const float* eb = emis + (long)b * T_ * C_;
  float alpha = valid ? (start_t[j] + eb[j]) : NEG;

  for (int t = 1; t < len; ++t) {
    // online logsumexp over predecessor states i
    float mx = NEG, sum = 0.0f;
#pragma unroll
    for (int i = 0; i < C_; ++i) {
      const float ai = __shfl(alpha, i, 32);
      const float s  = ai + tcol[i];
      if (s > mx) { sum = sum * expf(mx - s) + 1.0f; mx = s; }
      else        { sum += expf(s - mx); }
    }
    const float nv = mx + logf(sum) + (valid ? eb[t * C_ + j] : 0.0f);
    alpha = valid ? nv : NEG;
  }

  // log_z = logsumexp_j(alpha_j + end_trans_j)
  float v = valid ? (alpha + end_t[j]) : NEG;
  float mx = v;
#pragma unroll
  for (int off = 16; off; off >>= 1) mx = fmaxf(mx, __shfl_xor(mx, off, 32));
  float se = valid ? expf(v - mx) : 0.0f;
#pragma unroll
  for (int off = 16; off; off >>= 1) se += __shfl_xor(se, off, 32);
  const float lz = mx + logf(se);

  // gold path score, lanes stride over time
  const int* lb = labels + b * T_;
  float g = 0.0f;
  for (int t = j; t < len; t += 32) {
    g += eb[t * C_ + lb[t]];
    if (t >= 1) g += trans[lb[t - 1] * C_ + lb[t]];
  }
#pragma unroll
  for (int off = 16; off; off >>= 1) g += __shfl_xor(g, off, 32);

  if (j == 0) {
    g += start_t[lb[0]] + end_t[lb[len - 1]];
    loss_b[b] = lz - g;
  }
}

// ---------------------------------------------------------------------------
// Kernel 3: fixed-order reduction of per-batch losses -> deterministic scalar.
// ---------------------------------------------------------------------------
__global__ void reduce_loss_kernel(const float* __restrict__ loss_b,
                                   float* __restrict__ out) {
  if (threadIdx.x == 0) {
    float s = 0.0f;
    for (int i = 0; i < B_; ++i) s += loss_b[i];
    out[0] = s;
  }
}

extern "C" void kernel_launch(void* const* d_in, const int* in_sizes, int n_in,
                              void* d_out, int out_size, void* d_ws, size_t ws_size,
                              hipStream_t stream) {
  const float* feat   = (const float*)d_in[0];
  const float* W      = (const float*)d_in[1];
  const float* bias   = (const float*)d_in[2];
  const float* trans  = (const float*)d_in[3];
  const float* st     = (const float*)d_in[4];
  const float* et     = (const float*)d_in[5];
  const int*   lens   = (const int*)d_in[6];
  const int*   labels = (const int*)d_in[7];

  float* out    = (float*)d_out;
  float* emis   = out + 1;            // d_out = [loss, emissions(B,T,C)]
  float* loss_b = (float*)d_ws;       // 64 floats of scratch

  const int m_tiles = (B_ * T_) / 16;                 // 2048
  const int blocks  = m_tiles / WAVES_PER_BLOCK;      // 256

  emis_gemm_kernel<<<blocks, 256, 0, stream>>>(feat, W, bias, emis);
  crf_fwd_kernel<<<B_, 32, 0, stream>>>(emis, trans, st, et, lens, labels, loss_b);
  reduce_loss_kernel<<<1, 32, 0, stream>>>(loss_b, out);
}